// Integrator_50852412785439
// MI455X (gfx1250) — compile-verified
//
#include <hip/hip_runtime.h>
#include <hip/hip_bf16.h>

typedef float v2f __attribute__((ext_vector_type(2)));
typedef float v8f __attribute__((ext_vector_type(8)));

#define NB   16
#define HH   512
#define WW   512
#define HWP  (HH * WW)              // 262144 pixels per plane
#define NDISP (NB * 2 * HWP)        // 8388608 floats
#define NLD   (NB * HWP)            // 4194304 floats
#define EPS  0.0078125f             // 2^-7
#define SCAL (512.0f / 511.0f)      // S/(S-1) remap factor

// ---------------------------------------------------------------------------
// Sobel kernel coefficient for the A-matrix of the WMMA Jacobian GEMM.
// Row m (0..15): m<4 are the live outputs [dy*ch0, dx*ch0, dy*ch1, dx*ch1],
// rows 4..15 are zero padding. Tap t (0..19): t = c*9 + ky*3 + kx, t>=18 pad.
__device__ __forceinline__ float sobel_coef(int m, int t) {
    if (m >= 4 || t >= 18) return 0.0f;
    int c  = t / 9;
    int r  = t - c * 9;
    int ky = r / 3;
    int kx = r - ky * 3;
    int oc   = m >> 1;   // which input channel this output row differentiates
    int odir = m & 1;    // 0 = d/dy (dy kernel), 1 = d/dx (dx kernel)
    if (c != oc) return 0.0f;
    float dyv = (ky == 1) ? 0.0f : ((ky == 0 ? -1.0f : 1.0f) * ((kx == 1) ? 2.0f : 1.0f));
    float dxv = (kx == 1) ? 0.0f : ((kx == 0 ? -1.0f : 1.0f) * ((ky == 1) ? 2.0f : 1.0f));
    return 0.125f * (odir ? dxv : dyv);
}

// Replicate-padded tap gather for the B-matrix (im2col column for one pixel).
__device__ __forceinline__ float tapval(const float* __restrict__ vel,
                                        int n, int y, int x, int t) {
    if (t >= 18) return 0.0f;
    int c  = t / 9;
    int r  = t - c * 9;
    int ky = r / 3 - 1;
    int kx = r - (r / 3) * 3 - 1;
    int yy = min(max(y + ky, 0), HH - 1);
    int xx = min(max(x + kx, 0), WW - 1);
    return vel[(((size_t)n * 2 + c) * HH + yy) * WW + xx];
}

// ---------------------------------------------------------------------------
// Init kernel: disp = eps*vel  (thread-parallel)  and
// ldjac0 = -sum_{n=1..4} (-eps)^n tr(J^n)/n with J from the Sobel stencil,
// computed as 5 chained V_WMMA_F32_16X16X4_F32 (K = 18 taps padded to 20),
// 16 pixels per wave-iteration.
__global__ void __launch_bounds__(256)
init_kernel(const float* __restrict__ vel,
            float* __restrict__ disp,
            float* __restrict__ ld) {
    // --- part 1: disp = eps * vel (exact multiple of total threads) ---
    int tid      = blockIdx.x * blockDim.x + threadIdx.x;
    int nthreads = gridDim.x * blockDim.x;
    for (int i = tid; i < NDISP; i += nthreads)
        disp[i] = EPS * vel[i];

    // --- part 2: Jacobian GEMM + log-det series, 16 pixels per chunk ---
    int lane   = threadIdx.x & 31;
    int wave   = (blockIdx.x * blockDim.x + threadIdx.x) >> 5;
    int nwaves = nthreads >> 5;
    int m   = lane & 15;   // A-matrix row held by this lane
    int hi  = lane >> 4;   // K half (lanes 16..31 hold K = r+2)
    int npx = lane & 15;   // B/D column (pixel within chunk)

    const int NCHUNK = NB * HH * (WW / 16);   // 262144 chunks
    for (int chunk = wave; chunk < NCHUNK; chunk += nwaves) {
        int x0 = (chunk & 31) * 16;
        int y  = (chunk >> 5) & (HH - 1);
        int n  = chunk >> 14;
        int x  = x0 + npx;

        v8f acc = {0.f, 0.f, 0.f, 0.f, 0.f, 0.f, 0.f, 0.f};
        #pragma unroll
        for (int i = 0; i < 5; ++i) {
            int k0 = 4 * i + 2 * hi;   // tap index for VGPR 0 of A/B
            v2f a, b;
            a.x = sobel_coef(m, k0);
            a.y = sobel_coef(m, k0 + 1);
            b.x = tapval(vel, n, y, x, k0);
            b.y = tapval(vel, n, y, x, k0 + 1);
            acc = __builtin_amdgcn_wmma_f32_16x16x4_f32(
                false, a, false, b, (short)0, acc, false, false);
        }

        // Lanes 0..15: acc.s0..s3 = J00,J01,J10,J11 for pixel (n,y,x0+lane).
        float ja = acc.s0, jb = acc.s1, jc = acc.s2, jd = acc.s3;
        float xa = ja, xb = jb, xc = jc, xd = jd;     // X = J
        float neps = -EPS;
        float p  = neps;
        float lv = -(p * (xa + xd));                  // n = 1
        #pragma unroll
        for (int nn = 2; nn <= 4; ++nn) {             // X = X*J, accumulate
            float ta = xa * ja + xb * jc;
            float tb = xa * jb + xb * jd;
            float tc = xc * ja + xd * jc;
            float td = xc * jb + xd * jd;
            xa = ta; xb = tb; xc = tc; xd = td;
            p *= neps;
            lv -= p * (xa + xd) / (float)nn;
        }
        if (lane < 16)
            ld[(size_t)n * HWP + y * WW + x0 + lane] = lv;
        // (reconverged here; next iteration's WMMA sees full EXEC)
    }
}

// ---------------------------------------------------------------------------
// Bilinear samplers, zeros padding, pixel-coordinate inputs.
__device__ __forceinline__ float2 bilin2(const float* __restrict__ p0,
                                         const float* __restrict__ p1,
                                         float sx, float sy) {
    float xf = floorf(sx), yf = floorf(sy);
    float wx = sx - xf,    wy = sy - yf;
    int x0 = (int)xf, y0 = (int)yf;
    int x1 = x0 + 1,  y1 = y0 + 1;
    bool bx0 = ((unsigned)x0 < (unsigned)WW), bx1 = ((unsigned)x1 < (unsigned)WW);
    bool by0 = ((unsigned)y0 < (unsigned)HH), by1 = ((unsigned)y1 < (unsigned)HH);
    float w00 = (1.f - wx) * (1.f - wy), w01 = wx * (1.f - wy);
    float w10 = (1.f - wx) * wy,         w11 = wx * wy;
    int i00 = y0 * WW + x0, i01 = i00 + 1, i10 = i00 + WW, i11 = i10 + 1;
    float a0 = 0.f, a1 = 0.f;
    if (by0 && bx0) { a0 += w00 * p0[i00]; a1 += w00 * p1[i00]; }
    if (by0 && bx1) { a0 += w01 * p0[i01]; a1 += w01 * p1[i01]; }
    if (by1 && bx0) { a0 += w10 * p0[i10]; a1 += w10 * p1[i10]; }
    if (by1 && bx1) { a0 += w11 * p0[i11]; a1 += w11 * p1[i11]; }
    return make_float2(a0, a1);
}

__device__ __forceinline__ float bilin1(const float* __restrict__ p,
                                        float sx, float sy) {
    float xf = floorf(sx), yf = floorf(sy);
    float wx = sx - xf,    wy = sy - yf;
    int x0 = (int)xf, y0 = (int)yf;
    int x1 = x0 + 1,  y1 = y0 + 1;
    bool bx0 = ((unsigned)x0 < (unsigned)WW), bx1 = ((unsigned)x1 < (unsigned)WW);
    bool by0 = ((unsigned)y0 < (unsigned)HH), by1 = ((unsigned)y1 < (unsigned)HH);
    float w00 = (1.f - wx) * (1.f - wy), w01 = wx * (1.f - wy);
    float w10 = (1.f - wx) * wy,         w11 = wx * wy;
    int i00 = y0 * WW + x0, i01 = i00 + 1, i10 = i00 + WW, i11 = i10 + 1;
    float a = 0.f;
    if (by0 && bx0) a += w00 * p[i00];
    if (by0 && bx1) a += w01 * p[i01];
    if (by1 && bx0) a += w10 * p[i10];
    if (by1 && bx1) a += w11 * p[i11];
    return a;
}

// ---------------------------------------------------------------------------
// One scaling-and-squaring step (fused disp + ldjac update, ping-pong).
__global__ void __launch_bounds__(256)
step_kernel(const float* __restrict__ dispS, const float* __restrict__ ldS,
            float* __restrict__ dispD,       float* __restrict__ ldD) {
    int idx = blockIdx.x * blockDim.x + threadIdx.x;    // one thread per pixel
    int x = idx & (WW - 1);
    int y = (idx >> 9) & (HH - 1);
    int n = idx >> 18;

    const float* d0p = dispS + (size_t)n * 2 * HWP;     // y-channel plane
    const float* d1p = d0p + HWP;                       // x-channel plane
    int pix = y * WW + x;
    float d0 = d0p[pix];
    float d1 = d1p[pix];

    // warp disp by itself
    float sy = ((float)y + d0) * SCAL - 0.5f;
    float sx = ((float)x + d1) * SCAL - 0.5f;
    float2 s = bilin2(d0p, d1p, sx, sy);
    float nd0 = d0 + s.x;
    float nd1 = d1 + s.y;

    // warp ldjac by the NEW disp
    const float* ldp = ldS + (size_t)n * HWP;
    float lv  = ldp[pix];
    float sy2 = ((float)y + nd0) * SCAL - 0.5f;
    float sx2 = ((float)x + nd1) * SCAL - 0.5f;
    lv += bilin1(ldp, sx2, sy2);

    dispD[(size_t)n * 2 * HWP + pix]       = nd0;
    dispD[(size_t)n * 2 * HWP + HWP + pix] = nd1;
    ldD[(size_t)n * HWP + pix]             = lv;
}

// ---------------------------------------------------------------------------
extern "C" void kernel_launch(void* const* d_in, const int* in_sizes, int n_in,
                              void* d_out, int out_size, void* d_ws, size_t ws_size,
                              hipStream_t stream) {
    const float* vel = (const float*)d_in[0];
    // d_in[1] = logdet_jac flag (always 1 in this workload; we always compute it)

    // ping-pong buffer A in workspace, buffer B aliases d_out (same layout as
    // the required output: disp[NDISP] followed by ldjac[NLD]).
    float* dispA = (float*)d_ws;
    float* ldA   = dispA + NDISP;
    float* dispB = (float*)d_out;
    float* ldB   = dispB + NDISP;

    init_kernel<<<2048, 256, 0, stream>>>(vel, dispA, ldA);

    for (int s = 0; s < 7; ++s) {
        const float* ds = (s & 1) ? dispB : dispA;
        const float* ls = (s & 1) ? ldB   : ldA;
        float*       dd = (s & 1) ? dispA : dispB;
        float*       ld = (s & 1) ? ldA   : ldB;
        step_kernel<<<NLD / 256, 256, 0, stream>>>(ds, ls, dd, ld);
    }
    // s = 6 is even -> final result lands in d_out (dispB / ldB). 
}